// RPN_52647709115001
// MI455X (gfx1250) — compile-verified
//
#include <hip/hip_runtime.h>
#include <cstdint>
#include <cstddef>

// ---------------- constants (match reference) ----------------
constexpr int   B_IMG     = 16;
constexpr int   A_CNT     = 250000;
constexpr int   PRE_N     = 2000;
constexpr int   POST_N    = 1000;
constexpr float NMS_T     = 0.7f;
constexpr float NEG_INFF  = -1e9f;
constexpr float BBOX_CLIP = 4.135166556742356f;   // log(1000/16)

constexpr int NBINS     = 2048;   // top 11 bits of ordered key
constexpr int CAND_CAP  = 4096;
constexpr int TOPSTRIDE = 2048;   // padded row stride for top boxes/scores
constexpr int NWORDS    = 64;     // padded mask row: 64 words (word 63 forced to 0)
constexpr int ROWS      = 2016;   // mask rows actually written (63 row-blocks)

__device__ __forceinline__ unsigned ordered_key(float f) {
    unsigned u = __float_as_uint(f);
    return (u & 0x80000000u) ? ~u : (u | 0x80000000u);  // monotone: bigger float -> bigger key
}

// ---------------- kernel 1: per-image radix-select + gather ----------------
__global__ __launch_bounds__(256)
void rpn_select(const float* __restrict__ obj,
                unsigned long long* __restrict__ cand,  // [B][CAND_CAP]
                int* __restrict__ counts)               // [B]
{
    __shared__ int hist[NBINS];
    __shared__ int sT;
    __shared__ int ctr;
    const int b = blockIdx.x;
    const float* o = obj + (size_t)b * A_CNT;

    for (int i = threadIdx.x; i < NBINS; i += blockDim.x) hist[i] = 0;
    if (threadIdx.x == 0) ctr = 0;
    __syncthreads();

    for (int i = threadIdx.x; i < A_CNT; i += blockDim.x) {
        __builtin_prefetch(o + i + 4096, 0, 0);        // global_prefetch_b8
        unsigned k = ordered_key(o[i]);
        atomicAdd(&hist[k >> 21], 1);
    }
    __syncthreads();

    if (threadIdx.x == 0) {
        int cum = 0, T = 0;
        for (int bin = NBINS - 1; bin >= 0; --bin) {
            cum += hist[bin];
            if (cum >= PRE_N) { T = bin; break; }
        }
        sT = T;
    }
    __syncthreads();

    const int T = sT;
    unsigned long long* cb = cand + (size_t)b * CAND_CAP;
    for (int i = threadIdx.x; i < A_CNT; i += blockDim.x) {
        unsigned k = ordered_key(o[i]);
        if ((int)(k >> 21) >= T) {
            int pos = atomicAdd(&ctr, 1);
            if (pos < CAND_CAP) {
                // key sorts ascending == (score desc, index asc)  -> lax.top_k tie-break
                cb[pos] = ((unsigned long long)(~k) << 32) | (unsigned)i;
            }
        }
    }
    __syncthreads();
    if (threadIdx.x == 0) counts[b] = (ctr < CAND_CAP) ? ctr : CAND_CAP;
}

// ---------------- kernel 2: bitonic sort + decode/clip top-2000 ----------------
__global__ __launch_bounds__(1024)
void rpn_sort_decode(const unsigned long long* __restrict__ cand,
                     const int* __restrict__ counts,
                     const float* __restrict__ obj,
                     const float* __restrict__ deltas,   // [B][A][4]
                     const float* __restrict__ anchors,  // [A][4]
                     const int* __restrict__ image_h,
                     const int* __restrict__ image_w,
                     float* __restrict__ topBoxes,       // [B][TOPSTRIDE][4]
                     float* __restrict__ topScores)      // [B][TOPSTRIDE]
{
    __shared__ unsigned long long keys[CAND_CAP];
    const int b = blockIdx.x;
    const int n = counts[b];
    const unsigned long long* cb = cand + (size_t)b * CAND_CAP;

    for (int i = threadIdx.x; i < CAND_CAP; i += blockDim.x)
        keys[i] = (i < n) ? cb[i] : 0xFFFFFFFFFFFFFFFFull;
    __syncthreads();

    for (int k = 2; k <= CAND_CAP; k <<= 1) {
        for (int j = k >> 1; j > 0; j >>= 1) {
            for (int i = threadIdx.x; i < CAND_CAP; i += blockDim.x) {
                int p = i ^ j;
                if (p > i) {
                    bool asc = ((i & k) == 0);
                    unsigned long long a = keys[i], c = keys[p];
                    if ((a > c) == asc) { keys[i] = c; keys[p] = a; }
                }
            }
            __syncthreads();
        }
    }

    const float wmax = (float)image_w[0] - 1.0f;
    const float hmax = (float)image_h[0] - 1.0f;
    for (int i = threadIdx.x; i < PRE_N; i += blockDim.x) {
        unsigned idx = (unsigned)(keys[i] & 0xFFFFFFFFull);
        float score = obj[(size_t)b * A_CNT + idx];
        const float* an = anchors + (size_t)idx * 4;
        const float* de = deltas  + ((size_t)b * A_CNT + idx) * 4;
        float aw  = an[2] - an[0] + 1.0f;
        float ah  = an[3] - an[1] + 1.0f;
        float acx = an[0] + 0.5f * aw;
        float acy = an[1] + 0.5f * ah;
        float dw  = fminf(de[2], BBOX_CLIP);
        float dh  = fminf(de[3], BBOX_CLIP);
        float pcx = de[0] * aw + acx;
        float pcy = de[1] * ah + acy;
        float pw  = expf(dw) * aw;
        float ph  = expf(dh) * ah;
        float x1 = pcx - 0.5f * pw;
        float y1 = pcy - 0.5f * ph;
        float x2 = pcx + 0.5f * pw - 1.0f;
        float y2 = pcy + 0.5f * ph - 1.0f;
        x1 = fminf(fmaxf(x1, 0.0f), wmax);
        x2 = fminf(fmaxf(x2, 0.0f), wmax);
        y1 = fminf(fmaxf(y1, 0.0f), hmax);
        y2 = fminf(fmaxf(y2, 0.0f), hmax);
        float* tb = topBoxes + ((size_t)b * TOPSTRIDE + i) * 4;
        tb[0] = x1; tb[1] = y1; tb[2] = x2; tb[3] = y2;
        topScores[(size_t)b * TOPSTRIDE + i] = score;
    }
}

// ---------------- kernel 3: 32x32 IoU tile -> suppression bitmask ----------------
// One wave32 per tile. Column boxes staged into LDS via the CDNA5 async
// global->LDS data mover (ASYNCcnt path). grid.x == 64: the last column block
// has all j >= PRE_N, so it writes the zero padding word of each mask row.
__global__ __launch_bounds__(32)
void rpn_iou_mask(const float* __restrict__ topBoxes,   // [B][TOPSTRIDE][4]
                  unsigned* __restrict__ masks)         // [B][ROWS][NWORDS]
{
    __shared__ float colb[32 * 4];
    const int cblk = blockIdx.x;
    const int rblk = blockIdx.y;
    const int b    = blockIdx.z;
    const int lane = threadIdx.x;

    {
        // per-lane: copy one 16B column box straight into LDS (no VGPR data)
        unsigned lds_addr = (unsigned)(uintptr_t)(&colb[0]) + (unsigned)(lane * 16);
        unsigned voff     = (unsigned)(lane * 16);
        unsigned long long base =
            (unsigned long long)(uintptr_t)(topBoxes + ((size_t)b * TOPSTRIDE + (size_t)cblk * 32) * 4);
        asm volatile("global_load_async_to_lds_b128 %0, %1, %2 offset:0"
                     :: "v"(lds_addr), "v"(voff), "s"(base) : "memory");
        asm volatile("s_wait_asynccnt 0" ::: "memory");
    }
    __syncthreads();

    const int i = rblk * 32 + lane;
    const float* rb = topBoxes + ((size_t)b * TOPSTRIDE + i) * 4;
    float rx1 = rb[0], ry1 = rb[1], rx2 = rb[2], ry2 = rb[3];
    float rarea = (rx2 - rx1 + 1.0f) * (ry2 - ry1 + 1.0f);

    unsigned word = 0u;
    #pragma unroll 8
    for (int jj = 0; jj < 32; ++jj) {
        int j = cblk * 32 + jj;
        float cx1 = colb[jj * 4 + 0], cy1 = colb[jj * 4 + 1];
        float cx2 = colb[jj * 4 + 2], cy2 = colb[jj * 4 + 3];
        float carea = (cx2 - cx1 + 1.0f) * (cy2 - cy1 + 1.0f);
        float iw = fminf(rx2, cx2) - fmaxf(rx1, cx1) + 1.0f;
        float ih = fminf(ry2, cy2) - fmaxf(ry1, cy1) + 1.0f;
        iw = fmaxf(iw, 0.0f);
        ih = fmaxf(ih, 0.0f);
        float inter = iw * ih;
        float iou = inter / (rarea + carea - inter);
        if (iou > NMS_T && j < i && j < PRE_N) word |= (1u << jj);
    }
    masks[((size_t)b * ROWS + i) * NWORDS + cblk] = word;
}

// ---------------- kernel 4: greedy scan (async double-buffered rows) + pack ----
__global__ __launch_bounds__(32)
void rpn_nms_collect(const unsigned* __restrict__ masks,
                     const float* __restrict__ topBoxes,
                     const float* __restrict__ topScores,
                     float* __restrict__ out)            // [B][POST_N][5]
{
    __shared__ unsigned removed[NWORDS];
    __shared__ unsigned keepW[NWORDS];     // keep bitmask over the 2000 candidates
    __shared__ unsigned rowbuf[2][NWORDS]; // double-buffered mask rows
    __shared__ int wpre[NWORDS + 1];
    __shared__ int flag;
    const int b = blockIdx.x;
    const int lane = threadIdx.x;
    const unsigned* mbase = masks + (size_t)b * ROWS * NWORDS;

    removed[lane] = 0u;        removed[lane + 32] = 0u;
    keepW[lane]   = 0u;        keepW[lane + 32]   = 0u;
    __syncthreads();

    // prime the pipeline: async-stage row 0 (64 words = 8B per lane) into buf 0
    {
        unsigned lds_addr = (unsigned)(uintptr_t)(&rowbuf[0][0]) + (unsigned)(lane * 8);
        unsigned voff     = (unsigned)(lane * 8);
        unsigned long long base = (unsigned long long)(uintptr_t)mbase;
        asm volatile("global_load_async_to_lds_b64 %0, %1, %2 offset:0"
                     :: "v"(lds_addr), "v"(voff), "s"(base) : "memory");
    }

    for (int i = 0; i < PRE_N; ++i) {
        const int buf = i & 1;
        if (lane == 0) {
            int sup = (removed[i >> 5] >> (i & 31)) & 1;
            flag = sup;
            if (!sup) keepW[i >> 5] |= (1u << (i & 31));
        }
        // issue prefetch of the NEXT row while this row finishes / is consumed
        if (i + 1 < PRE_N) {
            unsigned lds_addr = (unsigned)(uintptr_t)(&rowbuf[buf ^ 1][0]) + (unsigned)(lane * 8);
            unsigned voff     = (unsigned)(lane * 8);
            unsigned long long base =
                (unsigned long long)(uintptr_t)(mbase + (size_t)(i + 1) * NWORDS);
            asm volatile("global_load_async_to_lds_b64 %0, %1, %2 offset:0"
                         :: "v"(lds_addr), "v"(voff), "s"(base) : "memory");
            // async loads complete in order: <=1 outstanding means row i has landed
            asm volatile("s_wait_asynccnt 1" ::: "memory");
        } else {
            asm volatile("s_wait_asynccnt 0" ::: "memory");
        }
        __syncthreads();
        if (!flag) {   // only surviving boxes may suppress (reference's dynamic keep)
            removed[lane]      |= rowbuf[buf][lane];
            removed[lane + 32] |= rowbuf[buf][lane + 32];
        }
        __syncthreads();
    }

    // word-level exclusive prefix of keep popcounts
    if (lane == 0) {
        int s = 0;
        for (int w = 0; w < NWORDS; ++w) { wpre[w] = s; s += __popc(keepW[w]); }
        wpre[NWORDS] = s;
    }
    __syncthreads();

    const int K = wpre[NWORDS];  // total kept
    // parallel pack: kept boxes (score order) to slots [0,K), suppressed
    // (index-ascending, score = NEG_INF) fill the remainder up to POST_N.
    float* ob = out + (size_t)b * POST_N * 5;
    for (int i = lane; i < PRE_N; i += 32) {
        const int w = i >> 5, bit = i & 31;
        const unsigned kw = keepW[w];
        const bool kp = (kw >> bit) & 1u;
        const unsigned lowmask = (bit == 0) ? 0u : (0xFFFFFFFFu >> (32 - bit));
        const int rank = wpre[w] + __popc(kw & lowmask);     // kept before i
        const int slot = kp ? rank : (K + (i - rank));       // i - rank = suppressed before i
        if (slot < POST_N) {
            const float* tb = topBoxes + ((size_t)b * TOPSTRIDE + i) * 4;
            float* o = ob + slot * 5;
            o[0] = tb[0]; o[1] = tb[1]; o[2] = tb[2]; o[3] = tb[3];
            o[4] = kp ? topScores[(size_t)b * TOPSTRIDE + i] : NEG_INFF;
        }
    }
}

// ---------------- launcher ----------------
extern "C" void kernel_launch(void* const* d_in, const int* in_sizes, int n_in,
                              void* d_out, int out_size, void* d_ws, size_t ws_size,
                              hipStream_t stream) {
    const float* obj     = (const float*)d_in[0];   // [16][250000]
    const float* deltas  = (const float*)d_in[1];   // [16][250000][4]
    const float* anchors = (const float*)d_in[2];   // [250000][4]
    const int*   img_h   = (const int*)d_in[3];     // scalar
    const int*   img_w   = (const int*)d_in[4];     // scalar

    char* ws = (char*)d_ws;
    auto aln = [](size_t x) { return (x + 255) & ~(size_t)255; };

    size_t off = 0;
    unsigned long long* cand = (unsigned long long*)(ws + off);
    off += aln((size_t)B_IMG * CAND_CAP * sizeof(unsigned long long));
    int* counts = (int*)(ws + off);
    off += aln((size_t)B_IMG * sizeof(int));
    float* topScores = (float*)(ws + off);
    off += aln((size_t)B_IMG * TOPSTRIDE * sizeof(float));
    float* topBoxes = (float*)(ws + off);
    off += aln((size_t)B_IMG * TOPSTRIDE * 4 * sizeof(float));
    unsigned* masks = (unsigned*)(ws + off);
    off += aln((size_t)B_IMG * ROWS * NWORDS * sizeof(unsigned));
    (void)off; (void)ws_size; (void)in_sizes; (void)n_in; (void)out_size;

    rpn_select<<<dim3(B_IMG), dim3(256), 0, stream>>>(obj, cand, counts);
    rpn_sort_decode<<<dim3(B_IMG), dim3(1024), 0, stream>>>(
        cand, counts, obj, deltas, anchors, img_h, img_w, topBoxes, topScores);
    rpn_iou_mask<<<dim3(NWORDS, 63, B_IMG), dim3(32), 0, stream>>>(topBoxes, masks);
    rpn_nms_collect<<<dim3(B_IMG), dim3(32), 0, stream>>>(masks, topBoxes, topScores,
                                                          (float*)d_out);
}